// CustomLoss_49057116455663
// MI455X (gfx1250) — compile-verified
//
#include <hip/hip_runtime.h>
#include <stdint.h>

typedef unsigned __int128 u128;
typedef float v2f __attribute__((ext_vector_type(2)));
typedef float v8f __attribute__((ext_vector_type(8)));

#define GH 10
#define GW 10
#define TPB 128   // 4 waves; 128 samples/block; 51,200 B LDS stage

// ---- compile-time bitboard masks (100-bit grid in u128) ----
constexpr u128 u128_one() { return (u128)1; }
constexpr u128 grid_mask() { return (u128_one() << 100) - 1; }
constexpr u128 col_mask(int c) {
  u128 m = 0;
  for (int r = 0; r < GH; ++r) m |= u128_one() << (r * 10 + c);
  return m;
}

// 32-lane f32 sum using the matrix pipe: D = A(16x4) * ones(4x16).
// A[m,0] = x(lane m), A[m,2] = x(lane m+16), A[*,1]=A[*,3]=0.
// => D[m,n] = x(m) + x(m+16) for every column n.
// Lane n<16 holds D rows 0..7 in its 8 VGPRs, lane n>=16 holds rows 8..15,
// so (per-lane sum of D) + (same from lane^16) = full wave sum.
__device__ __forceinline__ float wave_reduce_wmma(float x) {
  v2f a; a[0] = x;    a[1] = 0.0f;
  v2f b; b[0] = 1.0f; b[1] = 1.0f;
  v8f c = {0.f, 0.f, 0.f, 0.f, 0.f, 0.f, 0.f, 0.f};
  v8f d = __builtin_amdgcn_wmma_f32_16x16x4_f32(false, a, false, b,
                                                (short)0, c, false, false);
  float s = ((d[0] + d[1]) + (d[2] + d[3])) + ((d[4] + d[5]) + (d[6] + d[7]));
  s += __shfl_xor(s, 16, 32);
  return s;
}

__global__ void __launch_bounds__(TPB)
loss_kernel(const float* __restrict__ result, const int* __restrict__ points,
            float* __restrict__ bsums, int B) {
  __shared__ float smem[TPB * 100];
  __shared__ float wred[TPB / 32];

  const int tid = threadIdx.x;
  const int gid = blockIdx.x * TPB + tid;

  // ---- stage block's samples global -> LDS via CDNA5 async copies ----
  {
    const size_t blockBase = (size_t)blockIdx.x * (TPB * 400);
    const size_t maxOff = (size_t)B * 400 - 16;
    const uint32_t ldsBase = (uint32_t)(uintptr_t)(&smem[0]);
#pragma unroll
    for (int i = 0; i < 25; ++i) {
      const uint32_t ci = (uint32_t)(i * TPB + tid);   // 16B chunk id
      size_t go = blockBase + (size_t)ci * 16;
      if (go > maxOff) go = maxOff;                    // tail clamp (dup reads ok)
      const uint64_t gaddr = (uint64_t)(uintptr_t)result + go;
      const uint32_t laddr = ldsBase + ci * 16;
      asm volatile("global_load_async_to_lds_b128 %0, %1, off"
                   :: "v"(laddr), "v"(gaddr) : "memory");
    }
    asm volatile("s_wait_asynccnt 0x0" ::: "memory");
  }
  __syncthreads();

  const int s = (gid < B) ? gid : (B - 1);
  const int4 p = reinterpret_cast<const int4*>(points)[s];
  const int p0y = p.x, p0x = p.y, p1y = p.z, p1x = p.w;
  const float* r = &smem[tid * 100];

  const int base = abs(p0y - p1y) + abs(p0x - p1x);
  const float loss_start =
      (2.0f - (r[p0y * 10 + p0x] + r[p1y * 10 + p1x])) * 1000.0f;

  // ---- mask bitboard: round(r)==1 (round-half-even via rintf) ----
  uint32_t mw[4] = {0u, 0u, 0u, 0u};
#pragma unroll
  for (int k = 0; k < 50; ++k) {
    const float2 v = reinterpret_cast<const float2*>(r)[k];
    const int b0 = 2 * k, b1 = 2 * k + 1;
    if (rintf(v.x) == 1.0f) mw[b0 >> 5] |= 1u << (b0 & 31);
    if (rintf(v.y) == 1.0f) mw[b1 >> 5] |= 1u << (b1 & 31);
  }
  const u128 mask = ((u128)mw[3] << 96) | ((u128)mw[2] << 64) |
                    ((u128)mw[1] << 32) | (u128)mw[0];

  // ---- 8-connected flood fill, exactly H+W=20 dilations (matches ref) ----
  constexpr u128 GRID = grid_mask();
  constexpr u128 NOTCOL0 = GRID & ~col_mask(0);
  constexpr u128 NOTCOL9 = GRID & ~col_mask(9);
  u128 cl = u128_one() << (p0y * 10 + p0x);
#pragma unroll 1
  for (int it = 0; it < 20; ++it) {
    const u128 e  = (cl << 1) & NOTCOL0;
    const u128 w  = (cl >> 1) & NOTCOL9;
    const u128 ew = e | w;
    const u128 h  = cl | ew;
    const u128 nb = ew | (h << 10) | (h >> 10);
    cl |= nb & mask;
  }
  const float csz =
      (float)(__popcll((uint64_t)cl) + __popcll((uint64_t)(cl >> 64)));

  // column terms
  int cx01[10], ex[10];
#pragma unroll
  for (int j = 0; j < 10; ++j) {
    const int a0 = abs(j - p0x), a1 = abs(j - p1x);
    cx01[j] = a0 + a1;
    ex[j] = a1;
  }

  // ---- pass 1: manhattan-detour penalty + argmin(cluster dist to end) ----
  float single = 0.0f;
  int bm = GH * GW + 10, bidx = 0;
#pragma unroll 1
  for (int i = 0; i < 10; ++i) {
    const uint32_t rowb = (uint32_t)(cl >> (i * 10)) & 0x3FFu;
    const int ryb = abs(i - p0y) + abs(i - p1y) - base;
    const int ey = abs(i - p1y);
#pragma unroll
    for (int j = 0; j < 10; ++j) {
      const float rv = r[i * 10 + j];
      const int delta = ryb + cx01[j];
      const float fa = fmaf(rv, -20.0f, 20.0f);        // (1-rv)*CELL_W
      const float fb = rv * ((float)delta * 0.5f);
      single += (delta == 0) ? fa : fb;
      const bool cbit = (rowb >> j) & 1;
      const int de = ey + ex[j];
      const bool take = cbit && (de < bm);              // strict < => first-min
      bm = take ? de : bm;
      bidx = take ? (i * 10 + j) : bidx;
    }
  }

  // ---- pass 2: transposed membership (cluster.T), lonely + size pen ----
  float lonely = 0.0f, cpr = 0.0f;
#pragma unroll 1
  for (int a = 0; a < 10; ++a) {
    const uint32_t rowb = (uint32_t)(cl >> (a * 10)) & 0x3FFu;
#pragma unroll
    for (int b2 = 0; b2 < 10; ++b2) {
      const float rv = r[b2 * 10 + a];     // in_cl[i][j]=cluster[j][i]
      const bool ic = (rowb >> b2) & 1;
      lonely += ic ? fmaf(rv, -5.0f, 5.0f) : 15.0f * rv;
      cpr += ic ? rv : 0.0f;
    }
  }
  const float cluster_size_pen = cpr * (csz * 12.0f);

  // ---- gap-to-endpoint ----
  const bool better = bm < base;
  const int ny = better ? (bidx / 10) : p0y;
  const int nx = better ? (bidx % 10) : p0x;
  const int gap = min(base, bm);
  const int oy = p1y - ny, ox = p1x - nx;
  int by = ny, bx = nx, bg = gap;
  auto upd = [&](bool cond, int cy, int cx2) {
    const int d = abs(cy - p1y) + abs(cx2 - p1x);
    const bool t = cond && (d < bg);
    by = t ? cy : by; bx = t ? cx2 : bx; bg = t ? d : bg;
  };
  bool cn = ox < 0;
  upd(cn, ny, nx - 1);
  upd(cn && (ny != 0), ny - 1, nx - 1);
  upd(cn && (ny != GH - 1), ny + 1, nx - 1);
  cn = ox > 0;
  upd(cn, ny, nx + 1);
  upd(cn && (ny != 0), ny - 1, nx + 1);
  upd(cn && (ny != GH - 1), ny + 1, nx + 1);
  upd(oy < 0, ny - 1, nx);
  upd(oy > 0, ny + 1, nx);
  const int ncy = min(max(by, 0), GH - 1);
  const int ncx = min(max(bx, 0), GW - 1);
  const float gap_pen = (float)gap * 300.0f * (1.0f - r[ncy * 10 + ncx]);

  float loss = loss_start + lonely + single + cluster_size_pen + gap_pen;
  if (gid >= B) loss = 0.0f;

  // ---- deterministic block reduction: WMMA wave-reduce + tiny LDS tree ----
  const float wsum = wave_reduce_wmma(loss);
  if ((tid & 31) == 0) wred[tid >> 5] = wsum;
  __syncthreads();
  if (tid == 0)
    bsums[blockIdx.x] = (wred[0] + wred[1]) + (wred[2] + wred[3]);
}

__global__ void __launch_bounds__(32)
final_reduce(const float* __restrict__ bsums, float* __restrict__ out, int n) {
  const int lane = threadIdx.x;
  float s = 0.0f;
  for (int k = lane; k < n; k += 32) s += bsums[k];
  const float t = wave_reduce_wmma(s);
  if (lane == 0) out[0] = t;
}

extern "C" void kernel_launch(void* const* d_in, const int* in_sizes, int n_in,
                              void* d_out, int out_size, void* d_ws, size_t ws_size,
                              hipStream_t stream) {
  (void)n_in; (void)out_size; (void)ws_size;
  const float* result = (const float*)d_in[0];
  const int* points = (const int*)d_in[1];
  const int B = in_sizes[0] / 100;
  const int nblocks = (B + TPB - 1) / TPB;
  float* bsums = (float*)d_ws;
  loss_kernel<<<nblocks, TPB, 0, stream>>>(result, points, bsums, B);
  final_reduce<<<1, 32, 0, stream>>>(bsums, (float*)d_out, nblocks);
}